// RhoNN_58514634441373
// MI455X (gfx1250) — compile-verified
//
#include <hip/hip_runtime.h>

typedef float v2f __attribute__((ext_vector_type(2)));
typedef float v8f __attribute__((ext_vector_type(8)));

#define N_NODES 100000
#define N_EDGES 200000
#define D_NODE  128
#define D_EDGE  16
#define N_REL   3

#define ROW_BLOCKS (N_NODES / 16)   // 6250
#define EDGE_TILES (N_EDGES / 16)   // 12500

#define KA_GRID 512
#define KB_GRID 512

// WMMA helper: D = A(16x4) * B(4x16) + C, full f32 (V_WMMA_F32_16X16X4_F32)
__device__ __forceinline__ v8f wmma_f32(v2f a, v2f b, v8f c) {
    return __builtin_amdgcn_wmma_f32_16x16x4_f32(
        /*neg_a=*/false, a, /*neg_b=*/false, b,
        /*c_mod=*/(short)0, c, /*reuse_a=*/false, /*reuse_b=*/false);
}

// ---------------------------------------------------------------------------
// Kernel A: out = x @ W_self + b        (fully overwrites d_out each call)
// Persistent blocks: stage W_self in LDS once, grid-stride over 16-row blocks.
// ---------------------------------------------------------------------------
__global__ __launch_bounds__(128)
void kA_self(const float* __restrict__ x, const float* __restrict__ Wself,
             const float* __restrict__ bias, float* __restrict__ out) {
    __shared__ float sW[D_NODE * D_NODE];          // 64 KB
    {
        const float4* s = (const float4*)Wself;
        float4* d = (float4*)sW;
        for (int i = threadIdx.x; i < D_NODE * D_NODE / 4; i += 128) d[i] = s[i];
    }
    __syncthreads();

    const int wave = threadIdx.x >> 5;
    const int lane = threadIdx.x & 31;
    const int m    = lane & 15;          // A row / B,C col within tile
    const int half = lane >> 4;          // K-half select for A/B fragments

    for (int rowBlock = blockIdx.x * 4 + wave; rowBlock < ROW_BLOCKS;
         rowBlock += KA_GRID * 4) {
        // A fragments: row (rowBlock*16+m), elements k = 4*kk + 2*half + {0,1}
        v2f a[32];
        const float* xr = x + (size_t)(rowBlock * 16 + m) * D_NODE + half * 2;
#pragma unroll
        for (int kk = 0; kk < 32; ++kk)
            a[kk] = *(const v2f*)(xr + kk * 4);

        for (int n = 0; n < 8; ++n) {
            const int col = n * 16 + m;
            v8f c = {};
#pragma unroll
            for (int kk = 0; kk < 32; ++kk) {
                const int krow = kk * 4 + half * 2;
                v2f bf;
                bf.x = sW[krow * D_NODE + col];
                bf.y = sW[(krow + 1) * D_NODE + col];
                c = wmma_f32(a[kk], bf, c);
            }
            const float bv = bias[col];
#pragma unroll
            for (int vg = 0; vg < 8; ++vg) {
                const int M = vg + half * 8;   // C layout: VGPR g -> row g (+8 hi)
                out[(size_t)(rowBlock * 16 + M) * D_NODE + col] = c[vg] + bv;
            }
        }
    }
}

// ---------------------------------------------------------------------------
// Kernel B: per 16-edge tile:  msg = x[src] @ W_rel[r] + ea @ We_rel[r]
//           atomically scatter-add msg rows into out[dst].
// Persistent blocks: stage W_rel[r] once, grid-stride over edge tiles.
// grid = (KB_GRID, N_REL), block = 128 (4 waves).
// ---------------------------------------------------------------------------
__global__ __launch_bounds__(128)
void kB_edges(const float* __restrict__ x, const float* __restrict__ eattr,
              const float* __restrict__ Wrel, const float* __restrict__ Werel,
              const int* __restrict__ eidx, float* __restrict__ out) {
    __shared__ float sW[D_NODE * D_NODE];          // 64 KB: W_rel[r]
    const int r = blockIdx.y;
    {
        const float4* s = (const float4*)(Wrel + (size_t)r * D_NODE * D_NODE);
        float4* d = (float4*)sW;
        for (int i = threadIdx.x; i < D_NODE * D_NODE / 4; i += 128) d[i] = s[i];
    }
    __syncthreads();

    const int wave = threadIdx.x >> 5;
    const int lane = threadIdx.x & 31;
    const int m    = lane & 15;
    const int half = lane >> 4;

    const float* We = Werel + (size_t)r * D_EDGE * D_NODE;   // 8 KB, cache-hot

    for (int t = blockIdx.x * 4 + wave; t < EDGE_TILES; t += KB_GRID * 4) {
        const int e0 = t * 16;

        // src/dst for this lane's tile row (lanes 16..31 mirror 0..15)
        const int2 sd = ((const int2*)eidx)[(size_t)r * N_EDGES + e0 + m];

        // Gather A fragments of x[src] (resident: 64 VGPRs)
        v2f ax[32];
        const float* xr = x + (size_t)sd.x * D_NODE + half * 2;
#pragma unroll
        for (int kk = 0; kk < 32; ++kk)
            ax[kk] = *(const v2f*)(xr + kk * 4);

        // A fragments of edge_attr row (K = 16 -> 4 steps)
        v2f ae[4];
        const float* er = eattr + ((size_t)r * N_EDGES + e0 + m) * D_EDGE + half * 2;
#pragma unroll
        for (int kk = 0; kk < 4; ++kk)
            ae[kk] = *(const v2f*)(er + kk * 4);

        for (int n = 0; n < 8; ++n) {
            const int col = n * 16 + m;
            v8f c = {};
#pragma unroll
            for (int kk = 0; kk < 32; ++kk) {
                const int krow = kk * 4 + half * 2;
                v2f bf;
                bf.x = sW[krow * D_NODE + col];
                bf.y = sW[(krow + 1) * D_NODE + col];
                c = wmma_f32(ax[kk], bf, c);
            }
#pragma unroll
            for (int kk = 0; kk < 4; ++kk) {
                const int krow = kk * 4 + half * 2;
                v2f bf;
                bf.x = We[krow * D_NODE + col];
                bf.y = We[(krow + 1) * D_NODE + col];
                c = wmma_f32(ae[kk], bf, c);
            }
            // Scatter: C element (vg,lane) is row M = vg + 8*half, col = n*16+m
#pragma unroll
            for (int vg = 0; vg < 8; ++vg) {
                const int M = vg + half * 8;
                const int dstRow = __shfl(sd.y, M, 32);
                atomicAdd(out + (size_t)dstRow * D_NODE + col, c[vg]);
            }
        }
    }
}

// ---------------------------------------------------------------------------
// Kernel C: in-place ReLU over d_out (float4)
// ---------------------------------------------------------------------------
__global__ __launch_bounds__(256)
void kC_relu(float* __restrict__ out, int n4) {
    int i = blockIdx.x * 256 + threadIdx.x;
    if (i < n4) {
        float4 v = ((float4*)out)[i];
        v.x = fmaxf(v.x, 0.0f);
        v.y = fmaxf(v.y, 0.0f);
        v.z = fmaxf(v.z, 0.0f);
        v.w = fmaxf(v.w, 0.0f);
        ((float4*)out)[i] = v;
    }
}

extern "C" void kernel_launch(void* const* d_in, const int* in_sizes, int n_in,
                              void* d_out, int out_size, void* d_ws, size_t ws_size,
                              hipStream_t stream) {
    (void)in_sizes; (void)n_in; (void)d_ws; (void)ws_size;
    const float* x     = (const float*)d_in[0];   // (100000,128)
    const float* eattr = (const float*)d_in[1];   // (3,200000,16)
    const float* Wrel  = (const float*)d_in[2];   // (3,128,128)
    const float* Werel = (const float*)d_in[3];   // (3,16,128)
    const float* Wself = (const float*)d_in[4];   // (128,128)
    const float* bias  = (const float*)d_in[5];   // (128,)
    const int*   eidx  = (const int*)d_in[6];     // (3,200000,2)
    float* out = (float*)d_out;                   // (100000,128)

    // Stage A: out = x @ W_self + b   (persistent blocks over 6250 row-blocks)
    kA_self<<<dim3(KA_GRID), dim3(128), 0, stream>>>(x, Wself, bias, out);

    // Stage B: atomic scatter-add of all relational messages into out
    kB_edges<<<dim3(KB_GRID, N_REL), dim3(128), 0, stream>>>(
        x, eattr, Wrel, Werel, eidx, out);

    // Stage C: ReLU in place
    const int n4 = N_NODES * D_NODE / 4;
    kC_relu<<<dim3((n4 + 255) / 256), dim3(256), 0, stream>>>(out, n4);
}